// YOLOv8Loss_81947976007887
// MI455X (gfx1250) — compile-verified
//
#include <hip/hip_runtime.h>
#include <hip/hip_bf16.h>
#include <stdint.h>

// ---------------------------------------------------------------------------
// YOLOv8-style loss, MI455X (gfx1250).
// Memory-optimal decomposition: dense pass touches ONLY the 3 obj channels
// (6.4 MB of 96.8 MB), everything else is a <=6144-cell gather.
// CDNA5 paths: global_load_async_to_lds_b128 (+ s_wait_asynccnt) staging,
// v_wmma_f32_16x16x4_f32 used as a wave32 horizontal-sum engine.
// ---------------------------------------------------------------------------

#define USE_ASYNC_LDS 1

typedef __attribute__((ext_vector_type(2))) float v2f;
typedef __attribute__((ext_vector_type(8))) float v8f;

struct Entry { int pk; int mask; float cx, cy, w, h; };   // 24 B

// anchors pre-divided by stride: [scale][anchor]
__device__ __constant__ float c_aw[3][3] = {
    {0.5f,    0.75f,    1.625f},
    {0.75f,   1.5625f,  1.5f},
    {1.4375f, 1.9375f,  4.65625f}};
__device__ __constant__ float c_ah[3][3] = {
    {0.625f,  1.5f,     1.125f},
    {1.5f,    1.125f,   3.0f},
    {1.125f,  2.46875f, 4.0625f}};

__device__ inline float softplus0(float x) {             // bce(x, t=0)
  return fmaxf(x, 0.0f) + log1pf(expf(-fabsf(x)));
}

#if defined(__has_builtin)
#if __has_builtin(__builtin_amdgcn_s_wait_asynccnt)
#define WAIT_ASYNC0() __builtin_amdgcn_s_wait_asynccnt(0)
#endif
#endif
#ifndef WAIT_ASYNC0
#define WAIT_ASYNC0() asm volatile("s_wait_asynccnt 0" ::: "memory")
#endif

// ---- wave32 sum via V_WMMA_F32_16X16X4_F32 (B = ones -> row sums) ---------
__device__ inline float wave_reduce_sum(float x) {
#if defined(__has_builtin) && __has_builtin(__builtin_amdgcn_wmma_f32_16x16x4_f32)
  // A 16x4 f32 layout: lanes 0-15 {K0,K1}, lanes 16-31 {K2,K3}.
  // a = {x, 0}  =>  A[m][0]=x_m, A[m][2]=x_{m+16}.  B = all ones.
  // D[m][n] = x_m + x_{m+16}; lane n holds D[v][n] (v=0..7 rows low/high half).
  v2f a; a.x = x;    a.y = 0.0f;
  v2f b; b.x = 1.0f; b.y = 1.0f;
  v8f c = {};
  c = __builtin_amdgcn_wmma_f32_16x16x4_f32(false, a, false, b,
                                            (short)0, c, false, false);
  float s = ((c[0] + c[1]) + (c[2] + c[3])) + ((c[4] + c[5]) + (c[6] + c[7]));
  s += __shfl_xor(s, 16, 32);        // rows 0-7 half + rows 8-15 half
  return s;
#else
  for (int off = 16; off; off >>= 1) x += __shfl_xor(x, off, 32);
  return x;
#endif
}

// ---------------------------------------------------------------------------
// Kernel 0: zero the 32-float accumulator header in workspace.
//   acc[0..2]=counts(int) acc[3..5]=S0 acc[6..8]=boxsum
//   acc[9..11]=objA(Σ bce0-x) acc[12..14]=objB(Σ bce0) acc[15..17]=clssum
// ---------------------------------------------------------------------------
__global__ void k_init(float* acc) {
  if (threadIdx.x < 32) acc[threadIdx.x] = 0.0f;
}

// ---------------------------------------------------------------------------
// Kernel 1: target assignment + in-wave dedupe (T=32 == one wave32).
// grid = 3 scales * 64 batches, block = 32 threads (one target each).
// last-writer-wins == max target index; class mask = union over colliders.
// ---------------------------------------------------------------------------
__global__ void k_assign(const float* __restrict__ targets, float* acc,
                         Entry* __restrict__ entries) {
  const int s = blockIdx.x / 64;
  const int b = blockIdx.x % 64;
  const int t = threadIdx.x;                       // 0..31
  const float* tg = targets + (b * 32 + t) * 5;
  const int   cls = (int)tg[0];
  const float cx = tg[1], cy = tg[2], w = tg[3], h = tg[4];
  const int G = (s == 0) ? 80 : (s == 1 ? 40 : 20);

  float best_iou = -1.0f; int best = 0;
  #pragma unroll
  for (int a = 0; a < 3; ++a) {
    const float aw = c_aw[s][a], ah = c_ah[s][a];
    const float inter = fminf(w, aw) * fminf(h, ah);
    const float uni   = w * h + aw * ah - inter;
    const float iou   = inter / (uni + 1e-6f);
    if (iou > best_iou) { best_iou = iou; best = a; }   // first max wins
  }
  const int gx = (int)(cx * (float)G);
  const int gy = (int)(cy * (float)G);
  const int key = (best << 16) | (gy << 8) | gx;

  __shared__ int keys[32];
  __shared__ int clss[32];
  keys[t] = key; clss[t] = cls;
  __syncthreads();

  bool winner = true; int mask = 0;
  #pragma unroll 4
  for (int u = 0; u < 32; ++u) {
    if (keys[u] == key) {
      mask |= 1 << clss[u];
      if (u > t) winner = false;                   // a later writer exists
    }
  }
  if (winner) {
    const int pos = atomicAdd((int*)&acc[s], 1);
    Entry e; e.pk = (b << 18) | key; e.mask = mask;
    e.cx = cx; e.cy = cy; e.w = w; e.h = h;
    entries[s * 2048 + pos] = e;
  }
}

// ---------------------------------------------------------------------------
// Kernel 2: dense S0 = Σ softplus(x) over the 3 obj channels of one scale.
// One block = one 256-float4 tile, staged to LDS with async copy.
// Element counts are exact multiples of 1024 floats for all 3 scales.
// ---------------------------------------------------------------------------
__global__ void k_objbce(const float* __restrict__ pred, int Q /* G*G/4 */,
                         float* acc_S0) {
  __shared__ float4 buf[256];
  const int tid = threadIdx.x;
  const int i = blockIdx.x * 256 + tid;            // float4 idx in [B,3,Q]
  const int b = i / (3 * Q);
  const int r = i - b * (3 * Q);
  const int a = r / Q;
  const int q = r - a * Q;
  const int src4 = (b * 45 + a * 15 + 4) * Q + q;  // obj channel = a*15+4

#if USE_ASYNC_LDS
  {
    const unsigned lds_off = (unsigned)(uintptr_t)&buf[tid]; // addr[31:0] = LDS byte offset
    const unsigned voff    = (unsigned)src4 * 16u;           // byte offset from base
    asm volatile("global_load_async_to_lds_b128 %0, %1, %2"
                 :: "v"(lds_off), "v"(voff), "s"(pred)
                 : "memory");
    WAIT_ASYNC0();
  }
  const float4 v = buf[tid];                       // own slot, own wave: no barrier
#else
  const float4 v = ((const float4*)pred)[src4];
#endif

  float s = softplus0(v.x) + softplus0(v.y) + softplus0(v.z) + softplus0(v.w);
  s = wave_reduce_sum(s);                          // v_wmma_f32_16x16x4_f32 path

  __shared__ float wsum[8];
  const int wave = tid >> 5, lane = tid & 31;
  if (lane == 0) wsum[wave] = s;
  __syncthreads();
  if (tid == 0) {
    float tot = 0.0f;
    #pragma unroll
    for (int k = 0; k < 8; ++k) tot += wsum[k];
    atomicAdd(acc_S0, tot);
  }
}

// ---------------------------------------------------------------------------
// Kernel 3: gather the <=2048 object cells per scale; box / obj / cls terms.
// ---------------------------------------------------------------------------
__global__ void k_gather(const float* __restrict__ p0,
                         const float* __restrict__ p1,
                         const float* __restrict__ p2,
                         float* acc, const Entry* __restrict__ entries) {
  const int gid = blockIdx.x * 256 + threadIdx.x;
  if (gid >= 3 * 2048) return;
  const int s = gid >> 11, j = gid & 2047;
  const int n = ((const int*)acc)[s];
  if (j >= n) return;

  const Entry e = entries[s * 2048 + j];
  const int b  = (e.pk >> 18) & 63;
  const int a  = (e.pk >> 16) & 3;
  const int gy = (e.pk >> 8) & 255;
  const int gx = e.pk & 255;
  const int G  = (s == 0) ? 80 : (s == 1 ? 40 : 20);
  const float* p = (s == 0) ? p0 : (s == 1 ? p1 : p2);
  const long long GG = (long long)G * G;
  const long long base0 = (long long)(b * 45 + a * 15) * GG + (long long)gy * G + gx;

  const float box[4] = {e.cx, e.cy, e.w, e.h};
  float bs = 0.0f;
  #pragma unroll
  for (int c = 0; c < 4; ++c) {
    const float d = p[base0 + (long long)c * GG] - box[c];
    bs += d * d;
  }
  const float x  = p[base0 + 4 * GG];
  const float b0 = softplus0(x);
  float cs = 0.0f;
  #pragma unroll
  for (int c = 0; c < 10; ++c) {
    const float xc = p[base0 + (long long)(5 + c) * GG];
    const float tt = (float)((e.mask >> c) & 1);
    cs += fmaxf(xc, 0.0f) - xc * tt + log1pf(expf(-fabsf(xc)));
  }
  atomicAdd(&acc[6 + s],  bs);        // box Σ
  atomicAdd(&acc[9 + s],  b0 - x);    // Σ bce(x,1) at obj cells
  atomicAdd(&acc[12 + s], b0);        // Σ bce0 at obj cells (noobj correction)
  atomicAdd(&acc[15 + s], cs);        // cls Σ
}

// ---------------------------------------------------------------------------
// Kernel 4: finalize — combine per-scale terms; out = {total, box, obj, cls}.
// ---------------------------------------------------------------------------
__global__ void k_finalize(const float* acc, float* out) {
  if (threadIdx.x == 0 && blockIdx.x == 0) {
    const int cells[3] = {64 * 3 * 6400, 64 * 3 * 1600, 64 * 3 * 400};
    float box = 0.0f, obj = 0.0f, cls = 0.0f;
    for (int s = 0; s < 3; ++s) {
      const float nobj  = (float)((const int*)acc)[s];
      const float nno   = (float)cells[s] - nobj;
      box += acc[6 + s] / (4.0f * nobj);
      obj += acc[9 + s] / nobj + 0.5f * (acc[3 + s] - acc[12 + s]) / nno;
      cls += acc[15 + s] / (10.0f * nobj);
    }
    out[1] = box; out[2] = obj; out[3] = cls; out[0] = box + obj + cls;
  }
}

// ---------------------------------------------------------------------------
extern "C" void kernel_launch(void* const* d_in, const int* in_sizes, int n_in,
                              void* d_out, int out_size, void* d_ws, size_t ws_size,
                              hipStream_t stream) {
  (void)in_sizes; (void)n_in; (void)out_size; (void)ws_size;
  const float* pred0   = (const float*)d_in[0];   // [64,45,80,80]
  const float* pred1   = (const float*)d_in[1];   // [64,45,40,40]
  const float* pred2   = (const float*)d_in[2];   // [64,45,20,20]
  const float* targets = (const float*)d_in[3];   // [64,32,5]
  float* out = (float*)d_out;

  float* acc      = (float*)d_ws;                      // 32-float header
  Entry* entries  = (Entry*)((char*)d_ws + 256);       // 3*2048 entries = 144 KB

  k_init    <<<1, 32, 0, stream>>>(acc);
  k_assign  <<<3 * 64, 32, 0, stream>>>(targets, acc, entries);
  k_objbce  <<<1200, 256, 0, stream>>>(pred0, 1600, &acc[3]);   // 80*80/4
  k_objbce  <<<300,  256, 0, stream>>>(pred1, 400,  &acc[4]);   // 40*40/4
  k_objbce  <<<75,   256, 0, stream>>>(pred2, 100,  &acc[5]);   // 20*20/4
  k_gather  <<<24, 256, 0, stream>>>(pred0, pred1, pred2, acc, entries);
  k_finalize<<<1, 32, 0, stream>>>(acc, out);
}